// FeatureInteraction_41996190220732
// MI455X (gfx1250) — compile-verified
//
#include <hip/hip_runtime.h>

// CDNA5 / gfx1250: wave32, WMMA (not MFMA).
typedef float v2f __attribute__((ext_vector_type(2)));
typedef float v8f __attribute__((ext_vector_type(8)));

#define NFEAT      64
#define NDIM       128
#define LDS_STRIDE 132          // 128 + 4 pad: row stride 528B -> 4-bank rotate, conflict-free b64 loads
#define NPAIR      2016         // 64*63/2
#define NTHREADS   320          // 10 waves = 10 (ti<=tj) tiles of the 4x4 tile grid

__global__ __launch_bounds__(NTHREADS, 1)
void FeatureInteraction_41996190220732_kernel(const float* __restrict__ x,
                                              float* __restrict__ out) {
    // One workgroup per batch element. Stage the 64x128 fp32 slab (32KB) once.
    __shared__ float tile[NFEAT * LDS_STRIDE];

    const int b   = blockIdx.x;
    const int tid = threadIdx.x;
    const float* xb = x + (size_t)b * (NFEAT * NDIM);

    // ---- cooperative stage: coalesced global_load_b128 -> ds_store_b128 ----
    // 64 rows * 32 float4 = 2048 float4 chunks across 320 threads.
    for (int idx = tid; idx < NFEAT * (NDIM / 4); idx += NTHREADS) {
        const int row = idx >> 5;        // idx / 32
        const int c4  = idx & 31;        // idx % 32
        const float4 v = ((const float4*)xb)[idx];   // flat: row*128 + c4*4 == idx*4
        *(float4*)&tile[row * LDS_STRIDE + (c4 << 2)] = v;
    }
    __syncthreads();

    // ---- one wave per gram tile (ti,tj), ti<=tj, 4x4 tiling of 64x64 ----
    static const int TI[10] = {0,0,0,0,1,1,1,2,2,3};
    static const int TJ[10] = {0,1,2,3,1,2,3,2,3,3};
    const int w    = tid >> 5;           // wave id 0..9 (wave32)
    const int lane = tid & 31;
    const int ti = TI[w];
    const int tj = TJ[w];

    // v_wmma_f32_16x16x4_f32 fragment addressing:
    //   A (16x4): lane L<16 -> row M=L, K=0..1 ; lane L+16 -> row M=L, K=2..3
    //   B (4x16): lane L<16 -> col N=L, K=0..1 ; lane L+16 -> col N=L, K=2..3
    // With B = X^T both fragments are the same per-lane pattern over row-major X.
    const int lhalf = lane >> 4;         // 0 or 1
    const int lmod  = lane & 15;
    const float* arow = &tile[(ti * 16 + lmod) * LDS_STRIDE + 2 * lhalf];
    const float* brow = &tile[(tj * 16 + lmod) * LDS_STRIDE + 2 * lhalf];

    v8f acc = {0.f, 0.f, 0.f, 0.f, 0.f, 0.f, 0.f, 0.f};

    #pragma unroll
    for (int d0 = 0; d0 < NDIM; d0 += 4) {
        const v2f a  = *(const v2f*)(arow + d0);   // ds_load_b64, bank-conflict-free
        const v2f bm = *(const v2f*)(brow + d0);
        // 8 args: (neg_a, A, neg_b, B, c_mod, C, reuse_a, reuse_b)
        acc = __builtin_amdgcn_wmma_f32_16x16x4_f32(
            false, a, false, bm, (short)0, acc, false, false);
    }

    // ---- scatter strict upper triangle ----
    // D layout: VGPR v, lane L: M = v + 8*(L>>4), N = L&15
    float* ob = out + (size_t)b * NPAIR;
    const int j = tj * 16 + lmod;
    #pragma unroll
    for (int v = 0; v < 8; ++v) {
        const int i = ti * 16 + v + 8 * lhalf;
        if (i < j) {
            // row-major (i, j>i) flattened: i*(127-i)/2 + (j-i-1)
            ob[(i * (127 - i)) / 2 + (j - i - 1)] = acc[v];
        }
    }
}

extern "C" void kernel_launch(void* const* d_in, const int* in_sizes, int n_in,
                              void* d_out, int out_size, void* d_ws, size_t ws_size,
                              hipStream_t stream) {
    const float* x = (const float*)d_in[0];
    float* out = (float*)d_out;
    const int batches = in_sizes[0] / (NFEAT * NDIM);   // 4096
    FeatureInteraction_41996190220732_kernel<<<dim3(batches), dim3(NTHREADS), 0, stream>>>(x, out);
}